// GATLayer_31688268710363
// MI455X (gfx1250) — compile-verified
//
#include <hip/hip_runtime.h>

#define N_NODES 50000
#define N_EDGES 800000
#define DIN 128
#define DE  64
#define DOUT 128
#define DLIN (DIN + DE)      // 192, K of edge GEMM
#define DAPP (DIN + DOUT)    // 256, K of node GEMM

typedef __attribute__((ext_vector_type(2))) float v2f;
typedef __attribute__((ext_vector_type(8))) float v8f;

__device__ __forceinline__ v2f ld2(const float* p) {
    return *reinterpret_cast<const v2f*>(p);
}

// ---------------- zero workspace (mx, denom, z are atomic accumulators) ----
__global__ void zero_ws_kernel(float* __restrict__ p, int n) {
    int i = blockIdx.x * 256 + threadIdx.x;
    if (i < n) p[i] = 0.0f;
}

// ---------------- pass 1: per-edge attention logit + segment max -----------
// one wave (32 lanes) per edge; e = relu(W_attn . [h_src, h_dst, ef] + b)
__global__ __launch_bounds__(256) void attn_kernel(
    const float* __restrict__ nfeats, const float* __restrict__ efeats,
    const int* __restrict__ src, const int* __restrict__ dst,
    const float* __restrict__ W_attn, const float* __restrict__ b_attn,
    float* __restrict__ e_buf, unsigned int* __restrict__ mx)
{
    const int lane = threadIdx.x & 31;
    const int wv   = threadIdx.x >> 5;
    const int edge = blockIdx.x * 8 + wv;   // grid = E/8 exact

    const int s = src[edge];
    const int d = dst[edge];
    const float* hs = nfeats + (size_t)s * DIN;
    const float* hd = nfeats + (size_t)d * DIN;
    const float* ef = efeats + (size_t)edge * DE;

    float sum = 0.0f;
#pragma unroll
    for (int r = 0; r < 4; ++r) sum += hs[lane + 32*r] * W_attn[lane + 32*r];
#pragma unroll
    for (int r = 0; r < 4; ++r) sum += hd[lane + 32*r] * W_attn[DIN + lane + 32*r];
#pragma unroll
    for (int r = 0; r < 2; ++r) sum += ef[lane + 32*r] * W_attn[2*DIN + lane + 32*r];
#pragma unroll
    for (int off = 16; off > 0; off >>= 1) sum += __shfl_xor(sum, off, 32);

    if (lane == 0) {
        float e = fmaxf(sum + b_attn[0], 0.0f);   // relu => e >= 0
        e_buf[edge] = e;
        // e >= 0 so uint bit-pattern order == float order; mx zero-initialized
        atomicMax(&mx[d], __float_as_uint(e));
    }
}

// ---------------- pass 2: edge GEMM (WMMA f32) + softmax-weighted scatter --
// one wave per 16-edge tile: M16x128 = X(16x192) * W_lin^T(192x128)
__global__ __launch_bounds__(256) void edge_gemm_kernel(
    const float* __restrict__ nfeats, const float* __restrict__ efeats,
    const int* __restrict__ src, const int* __restrict__ dst,
    const float* __restrict__ W_lin,
    const float* __restrict__ e_buf, const unsigned int* __restrict__ mx,
    float* __restrict__ denom, float* __restrict__ z)
{
    const int lane = threadIdx.x & 31;
    const int wv   = threadIdx.x >> 5;
    const int lo   = lane & 15;        // A: row M / B: col N
    const int hi   = lane >> 4;        // selects K pair within the 4-wide step
    const int tile = blockIdx.x * 8 + wv;     // grid = E/128 exact
    const int eBase = tile * 16;

    const int myEdge = eBase + lo;
    const int mySrc  = src[myEdge];
    const float* hs = nfeats + (size_t)mySrc * DIN;
    const float* ep = efeats + (size_t)myEdge * DE;

    v8f acc[8];
#pragma unroll
    for (int n = 0; n < 8; ++n) acc[n] = (v8f)0.0f;

    // K = 0..127 : h_src channels
    for (int kk = 0; kk < 32; ++kk) {
        const int k = kk * 4 + hi * 2;
        v2f a = ld2(hs + k);
#pragma unroll
        for (int n = 0; n < 8; ++n) {
            v2f b = ld2(W_lin + (size_t)(n * 16 + lo) * DLIN + k);
            acc[n] = __builtin_amdgcn_wmma_f32_16x16x4_f32(
                false, a, false, b, (short)0, acc[n], false, false);
        }
    }
    // K = 128..191 : edge-feature channels
    for (int kk = 32; kk < 48; ++kk) {
        const int k = kk * 4 + hi * 2;
        v2f a = ld2(ep + (k - DIN));
#pragma unroll
        for (int n = 0; n < 8; ++n) {
            v2f b = ld2(W_lin + (size_t)(n * 16 + lo) * DLIN + k);
            acc[n] = __builtin_amdgcn_wmma_f32_16x16x4_f32(
                false, a, false, b, (short)0, acc[n], false, false);
        }
    }

    // softmax numerator for this wave's 16 edges (lanes 0..15), denom atomic
    float exv = 0.0f;
    int   dn  = 0;
    if (lane < 16) {
        const int d   = dst[myEdge];
        const float ev = e_buf[myEdge];
        const float mv = __uint_as_float(mx[d]);
        exv = expf(ev - mv);
        dn  = d;
        atomicAdd(&denom[d], exv);
    }

    // scatter ex * m into z[dst]; D layout: acc[n][j] -> row j+8*hi, col n*16+lo
#pragma unroll
    for (int j = 0; j < 8; ++j) {
        const int   row = j + 8 * hi;
        const float exr = __shfl(exv, row, 32);
        const int   dr  = __shfl(dn,  row, 32);
        float* zp = z + (size_t)dr * DOUT + lo;
#pragma unroll
        for (int n = 0; n < 8; ++n) {
            atomicAdd(zp + n * 16, exr * acc[n][j]);
        }
    }
}

// ---------------- pass 3: node GEMM (WMMA f32) + bias + relu ---------------
// out(16x128) = relu([nfeat | z/denom](16x256) * W_apply^T + b)
__global__ __launch_bounds__(256) void node_out_kernel(
    const float* __restrict__ nfeats,
    const float* __restrict__ W_apply, const float* __restrict__ b_apply,
    const float* __restrict__ denom, const float* __restrict__ z,
    float* __restrict__ out)
{
    const int lane = threadIdx.x & 31;
    const int wv   = threadIdx.x >> 5;
    const int lo   = lane & 15;
    const int hi   = lane >> 4;
    const int tile = blockIdx.x * 8 + wv;
    if (tile >= (N_NODES / 16)) return;        // wave-uniform guard (3125 tiles)
    const int nodeBase = tile * 16;

    const int myNode = nodeBase + lo;
    const float* hn = nfeats + (size_t)myNode * DIN;
    const float* zn = z      + (size_t)myNode * DOUT;
    const float dnv = denom[myNode];
    const float rd  = dnv > 0.0f ? 1.0f / dnv : 1.0f;   // alpha normalization

    v8f acc[8];
#pragma unroll
    for (int n = 0; n < 8; ++n) acc[n] = (v8f)0.0f;

    // K = 0..127 : nfeats
    for (int kk = 0; kk < 32; ++kk) {
        const int k = kk * 4 + hi * 2;
        v2f a = ld2(hn + k);
#pragma unroll
        for (int n = 0; n < 8; ++n) {
            v2f b = ld2(W_apply + (size_t)(n * 16 + lo) * DAPP + k);
            acc[n] = __builtin_amdgcn_wmma_f32_16x16x4_f32(
                false, a, false, b, (short)0, acc[n], false, false);
        }
    }
    // K = 128..255 : z normalized by denom on the fly
    for (int kk = 32; kk < 64; ++kk) {
        const int k = kk * 4 + hi * 2;
        v2f a = ld2(zn + (k - DIN));
        a.x *= rd; a.y *= rd;
#pragma unroll
        for (int n = 0; n < 8; ++n) {
            v2f b = ld2(W_apply + (size_t)(n * 16 + lo) * DAPP + k);
            acc[n] = __builtin_amdgcn_wmma_f32_16x16x4_f32(
                false, a, false, b, (short)0, acc[n], false, false);
        }
    }

    float bias[8];
#pragma unroll
    for (int n = 0; n < 8; ++n) bias[n] = b_apply[n * 16 + lo];

#pragma unroll
    for (int j = 0; j < 8; ++j) {
        const int row = nodeBase + j + 8 * hi;
        float* op = out + (size_t)row * DOUT + lo;
#pragma unroll
        for (int n = 0; n < 8; ++n) {
            op[n * 16] = fmaxf(acc[n][j] + bias[n], 0.0f);
        }
    }
}

extern "C" void kernel_launch(void* const* d_in, const int* in_sizes, int n_in,
                              void* d_out, int out_size, void* d_ws, size_t ws_size,
                              hipStream_t stream) {
    const float* nfeats  = (const float*)d_in[0];
    const float* efeats  = (const float*)d_in[1];
    const int*   src     = (const int*)  d_in[2];
    const int*   dst     = (const int*)  d_in[3];
    const float* W_lin   = (const float*)d_in[4];
    const float* W_attn  = (const float*)d_in[5];
    const float* b_attn  = (const float*)d_in[6];
    const float* W_apply = (const float*)d_in[7];
    const float* b_apply = (const float*)d_in[8];
    float* out = (float*)d_out;

    // workspace layout (floats): e_buf[E] | mx[N] | denom[N] | z[N*128]
    float*        e_buf = (float*)d_ws;
    unsigned int* mx    = (unsigned int*)(e_buf + N_EDGES);
    float*        denom = (float*)(mx + N_NODES);
    float*        zbuf  = denom + N_NODES;

    const int zero_n = N_NODES * (DOUT + 2);   // mx + denom + z, contiguous
    zero_ws_kernel<<<(zero_n + 255) / 256, 256, 0, stream>>>((float*)mx, zero_n);

    attn_kernel<<<N_EDGES / 8, 256, 0, stream>>>(
        nfeats, efeats, src, dst, W_attn, b_attn, e_buf, mx);

    edge_gemm_kernel<<<N_EDGES / 128, 256, 0, stream>>>(
        nfeats, efeats, src, dst, W_lin, e_buf, mx, denom, zbuf);

    node_out_kernel<<<(N_NODES / 16 + 7) / 8, 256, 0, stream>>>(
        nfeats, W_apply, b_apply, denom, zbuf, out);
}